// GAT_70755291235031
// MI455X (gfx1250) — compile-verified
//
#include <hip/hip_runtime.h>
#include <hip/hip_bf16.h>

typedef __attribute__((ext_vector_type(2))) float v2f;
typedef __attribute__((ext_vector_type(8))) float v8f;

#define NEG_SLOPE 0.2f

// ---- monotone key encoding for float atomic max via u32 atomics ----
__device__ __forceinline__ unsigned enc_key(float f) {
    unsigned b = __float_as_uint(f);
    return (b & 0x80000000u) ? ~b : (b | 0x80000000u);
}
__device__ __forceinline__ float dec_key(unsigned k) {
    unsigned b = (k & 0x80000000u) ? (k ^ 0x80000000u) : ~k;
    return __uint_as_float(b);
}

// =================== GEMM1: h1[N,64] = feat[N,256] @ W1[256,64] ===================
// Block: 256 threads = 8 waves. Each wave owns a 16-row strip and computes all
// four 16-col tiles (4 accumulators) -> one A load feeds 4 WMMAs per k-step.
// Block covers 128 rows. W1 (64KB fp32) staged fully in LDS.
__global__ __launch_bounds__(256) void gat_gemm1(const float* __restrict__ feat,
                                                 const float* __restrict__ W1,
                                                 float* __restrict__ h1, int n) {
    __shared__ float ldsW[256 * 64];
    for (int i = threadIdx.x; i < (256 * 64) / 4; i += 256)
        ((float4*)ldsW)[i] = ((const float4*)W1)[i];
    __syncthreads();

    const int wave = threadIdx.x >> 5;
    const int lane = threadIdx.x & 31;
    const int row0 = blockIdx.x * 128 + wave * 16;
    const int half = lane >> 4;     // 0: K pair {0,1}, 1: K pair {2,3}
    const int m    = lane & 15;

    int arow_idx = row0 + m; if (arow_idx >= n) arow_idx = n - 1;
    const float* arow = feat + (size_t)arow_idx * 256;

    v8f acc0 = {}, acc1 = {}, acc2 = {}, acc3 = {};
    #pragma unroll 4
    for (int k = 0; k < 256; k += 4) {
        v2f a;
        a.x = arow[k + 2 * half];
        a.y = arow[k + 2 * half + 1];
        const float* wrow0 = ldsW + (k + 2 * half)     * 64 + m;
        const float* wrow1 = ldsW + (k + 2 * half + 1) * 64 + m;
        v2f b0; b0.x = wrow0[0];  b0.y = wrow1[0];
        v2f b1; b1.x = wrow0[16]; b1.y = wrow1[16];
        v2f b2; b2.x = wrow0[32]; b2.y = wrow1[32];
        v2f b3; b3.x = wrow0[48]; b3.y = wrow1[48];
        acc0 = __builtin_amdgcn_wmma_f32_16x16x4_f32(false, a, false, b0, (short)0, acc0, false, false);
        acc1 = __builtin_amdgcn_wmma_f32_16x16x4_f32(false, a, false, b1, (short)0, acc1, false, false);
        acc2 = __builtin_amdgcn_wmma_f32_16x16x4_f32(false, a, false, b2, (short)0, acc2, false, false);
        acc3 = __builtin_amdgcn_wmma_f32_16x16x4_f32(false, a, false, b3, (short)0, acc3, false, false);
    }

    if (row0 + 15 < n) {            // fast path: full tile, straight-line stores
        #pragma unroll
        for (int r = 0; r < 8; ++r) {
            float* orow = h1 + (size_t)(row0 + r + 8 * half) * 64 + m;
            orow[0] = acc0[r]; orow[16] = acc1[r]; orow[32] = acc2[r]; orow[48] = acc3[r];
        }
    } else {
        #pragma unroll
        for (int r = 0; r < 8; ++r) {
            int row = row0 + r + 8 * half;
            if (row < n) {
                float* orow = h1 + (size_t)row * 64 + m;
                orow[0] = acc0[r]; orow[16] = acc1[r]; orow[32] = acc2[r]; orow[48] = acc3[r];
            }
        }
    }
}

// =================== GEMM2: lin2[N,64pad] = x[N,64] @ W2pad[64,64] ===================
__global__ __launch_bounds__(256) void gat_gemm2(const float* __restrict__ x,
                                                 const float* __restrict__ W2,
                                                 float* __restrict__ lin2, int n) {
    __shared__ float ldsW[64 * 64];
    for (int i = threadIdx.x; i < 64 * 64; i += 256) {
        int kk = i >> 6, cc = i & 63;
        ldsW[i] = (cc < 50) ? W2[kk * 50 + cc] : 0.0f;
    }
    __syncthreads();

    const int wave = threadIdx.x >> 5;
    const int lane = threadIdx.x & 31;
    const int row0 = blockIdx.x * 128 + wave * 16;
    const int half = lane >> 4;
    const int m    = lane & 15;

    int arow_idx = row0 + m; if (arow_idx >= n) arow_idx = n - 1;
    const float* arow = x + (size_t)arow_idx * 64;

    v8f acc0 = {}, acc1 = {}, acc2 = {}, acc3 = {};
    #pragma unroll 4
    for (int k = 0; k < 64; k += 4) {
        v2f a;
        a.x = arow[k + 2 * half];
        a.y = arow[k + 2 * half + 1];
        const float* wrow0 = ldsW + (k + 2 * half)     * 64 + m;
        const float* wrow1 = ldsW + (k + 2 * half + 1) * 64 + m;
        v2f b0; b0.x = wrow0[0];  b0.y = wrow1[0];
        v2f b1; b1.x = wrow0[16]; b1.y = wrow1[16];
        v2f b2; b2.x = wrow0[32]; b2.y = wrow1[32];
        v2f b3; b3.x = wrow0[48]; b3.y = wrow1[48];
        acc0 = __builtin_amdgcn_wmma_f32_16x16x4_f32(false, a, false, b0, (short)0, acc0, false, false);
        acc1 = __builtin_amdgcn_wmma_f32_16x16x4_f32(false, a, false, b1, (short)0, acc1, false, false);
        acc2 = __builtin_amdgcn_wmma_f32_16x16x4_f32(false, a, false, b2, (short)0, acc2, false, false);
        acc3 = __builtin_amdgcn_wmma_f32_16x16x4_f32(false, a, false, b3, (short)0, acc3, false, false);
    }

    if (row0 + 15 < n) {
        #pragma unroll
        for (int r = 0; r < 8; ++r) {
            float* orow = lin2 + (size_t)(row0 + r + 8 * half) * 64 + m;
            orow[0] = acc0[r]; orow[16] = acc1[r]; orow[32] = acc2[r]; orow[48] = acc3[r];
        }
    } else {
        #pragma unroll
        for (int r = 0; r < 8; ++r) {
            int row = row0 + r + 8 * half;
            if (row < n) {
                float* orow = lin2 + (size_t)row * 64 + m;
                orow[0] = acc0[r]; orow[16] = acc1[r]; orow[32] = acc2[r]; orow[48] = acc3[r];
            }
        }
    }
}

// =================== attention coefficients, layer 1: per (node, head) ===================
__global__ void gat_attn1(const float* __restrict__ h1, const float* __restrict__ att_s,
                          const float* __restrict__ att_d,
                          float* __restrict__ a_s, float* __restrict__ a_d, int n) {
    int idx = blockIdx.x * blockDim.x + threadIdx.x;
    if (idx >= n * 8) return;
    int node = idx >> 3, h = idx & 7;
    const float* row = h1 + (size_t)node * 64 + h * 8;
    float ss = 0.f, dd = 0.f;
    #pragma unroll
    for (int c = 0; c < 8; ++c) {
        float v = row[c];
        ss += v * att_s[h * 8 + c];
        dd += v * att_d[h * 8 + c];
    }
    a_s[idx] = ss; a_d[idx] = dd;
}

// =================== attention coefficients, layer 2: per node (1 head, 50 ch) ===================
__global__ void gat_attn2(const float* __restrict__ lin2, const float* __restrict__ att_s,
                          const float* __restrict__ att_d,
                          float* __restrict__ a_s, float* __restrict__ a_d, int n) {
    int i = blockIdx.x * blockDim.x + threadIdx.x;
    if (i >= n) return;
    const float* row = lin2 + (size_t)i * 64;
    float ss = 0.f, dd = 0.f;
    for (int c = 0; c < 50; ++c) {
        float v = row[c];
        ss += v * att_s[c];
        dd += v * att_d[c];
    }
    a_s[i] = ss; a_d[i] = dd;
}

// =================== edge segment-max, layer 1: per (edge, head) ===================
__global__ void gat_edge_max1(const int* __restrict__ ei, int e, int etot,
                              const float* __restrict__ a_s, const float* __restrict__ a_d,
                              unsigned* __restrict__ m1) {
    int idx = blockIdx.x * blockDim.x + threadIdx.x;
    if (idx >= etot * 8) return;
    int eidx = idx >> 3, h = idx & 7;
    int src, dst;
    if (eidx < e) { src = ei[eidx]; dst = ei[e + eidx]; }
    else          { src = dst = eidx - e; }
    float v = a_s[src * 8 + h] + a_d[dst * 8 + h];
    v = (v > 0.f) ? v : NEG_SLOPE * v;
    atomicMax(&m1[dst * 8 + h], enc_key(v));
}

// =================== edge exp-sum + weighted aggregation, layer 1: 1 wave / edge ===================
__global__ __launch_bounds__(256) void gat_edge_agg1(const int* __restrict__ ei, int e, int etot,
                                                     const float* __restrict__ a_s,
                                                     const float* __restrict__ a_d,
                                                     const unsigned* __restrict__ m1,
                                                     float* __restrict__ s1,
                                                     const float* __restrict__ h1,
                                                     float* __restrict__ agg1) {
    int eidx = blockIdx.x * 8 + (threadIdx.x >> 5);
    if (eidx >= etot) return;
    int lane = threadIdx.x & 31;
    int src, dst;
    if (eidx < e) { src = ei[eidx]; dst = ei[e + eidx]; }
    else          { src = dst = eidx - e; }
    #pragma unroll
    for (int half = 0; half < 2; ++half) {
        int hc = lane + 32 * half;
        int h  = hc >> 3;
        float v = a_s[src * 8 + h] + a_d[dst * 8 + h];
        v = (v > 0.f) ? v : NEG_SLOPE * v;
        float p = __expf(v - dec_key(m1[dst * 8 + h]));
        if ((hc & 7) == 0) atomicAdd(&s1[dst * 8 + h], p);
        atomicAdd(&agg1[(size_t)dst * 64 + hc], p * h1[(size_t)src * 64 + hc]);
    }
}

// =================== finalize layer 1: x = ELU(agg/s + b1), in place ===================
__global__ void gat_finalize1(float* __restrict__ agg1, const float* __restrict__ s1,
                              const float* __restrict__ b1, int n) {
    int idx = blockIdx.x * blockDim.x + threadIdx.x;
    if (idx >= n * 64) return;
    int node = idx >> 6, hc = idx & 63, h = hc >> 3;
    float v = agg1[idx] / s1[node * 8 + h] + b1[hc];
    agg1[idx] = (v > 0.f) ? v : (__expf(v) - 1.0f);
}

// =================== edge segment-max, layer 2: per edge ===================
__global__ void gat_edge_max2(const int* __restrict__ ei, int e, int etot,
                              const float* __restrict__ a_s, const float* __restrict__ a_d,
                              unsigned* __restrict__ m2) {
    int eidx = blockIdx.x * blockDim.x + threadIdx.x;
    if (eidx >= etot) return;
    int src, dst;
    if (eidx < e) { src = ei[eidx]; dst = ei[e + eidx]; }
    else          { src = dst = eidx - e; }
    float v = a_s[src] + a_d[dst];
    v = (v > 0.f) ? v : NEG_SLOPE * v;
    atomicMax(&m2[dst], enc_key(v));
}

// =================== edge exp-sum + aggregation, layer 2: 1 wave / edge ===================
__global__ __launch_bounds__(256) void gat_edge_agg2(const int* __restrict__ ei, int e, int etot,
                                                     const float* __restrict__ a_s,
                                                     const float* __restrict__ a_d,
                                                     const unsigned* __restrict__ m2,
                                                     float* __restrict__ s2,
                                                     const float* __restrict__ lin2,
                                                     float* __restrict__ agg2) {
    int eidx = blockIdx.x * 8 + (threadIdx.x >> 5);
    if (eidx >= etot) return;
    int lane = threadIdx.x & 31;
    int src, dst;
    if (eidx < e) { src = ei[eidx]; dst = ei[e + eidx]; }
    else          { src = dst = eidx - e; }
    float v = a_s[src] + a_d[dst];
    v = (v > 0.f) ? v : NEG_SLOPE * v;
    float p = __expf(v - dec_key(m2[dst]));
    if (lane == 0) atomicAdd(&s2[dst], p);
    #pragma unroll
    for (int half = 0; half < 2; ++half) {
        int c = lane + 32 * half;
        if (c < 50)
            atomicAdd(&agg2[(size_t)dst * 64 + c], p * lin2[(size_t)src * 64 + c]);
    }
}

// =================== finalize layer 2: scores = agg2/s2 + b2 ===================
__global__ void gat_finalize2(float* __restrict__ agg2, const float* __restrict__ s2,
                              const float* __restrict__ b2, int n) {
    int idx = blockIdx.x * blockDim.x + threadIdx.x;
    if (idx >= n * 64) return;
    int c = idx & 63;
    if (c >= 50) return;
    agg2[idx] = agg2[idx] / s2[idx >> 6] + b2[c];
}

// =================== loss / pred / labels ===================
__global__ void gat_loss(const float* __restrict__ scores, const long long* __restrict__ label,
                         float* __restrict__ out, int n) {
    int i = blockIdx.x * blockDim.x + threadIdx.x;
    if (i >= n) return;
    const float* row = scores + (size_t)i * 64;
    float mx = row[0]; int am = 0;
    for (int c = 1; c < 50; ++c) { float v = row[c]; if (v > mx) { mx = v; am = c; } }
    float se = 0.f;
    for (int c = 0; c < 50; ++c) se += __expf(row[c] - mx);
    float lse = mx + __logf(se);
    long long lb = label[i];
    float li = lse - row[(int)lb];
    atomicAdd(out, li / (float)n);        // mean NLL
    out[1 + i]     = (float)am;           // pred
    out[1 + n + i] = (float)lb;           // labels_m (mask all-true)
}

extern "C" void kernel_launch(void* const* d_in, const int* in_sizes, int n_in,
                              void* d_out, int out_size, void* d_ws, size_t ws_size,
                              hipStream_t stream) {
    const float*     feat   = (const float*)d_in[1];
    const int*       ei     = (const int*)d_in[2];
    const long long* label  = (const long long*)d_in[4];
    const float*     W1     = (const float*)d_in[5];
    const float*     att_s1 = (const float*)d_in[6];
    const float*     att_d1 = (const float*)d_in[7];
    const float*     b1     = (const float*)d_in[8];
    const float*     W2     = (const float*)d_in[9];
    const float*     att_s2 = (const float*)d_in[10];
    const float*     att_d2 = (const float*)d_in[11];
    const float*     b2     = (const float*)d_in[12];

    const int n    = in_sizes[0];
    const int e    = in_sizes[2] / 2;
    const int etot = e + n;
    const size_t nn = (size_t)n;

    // workspace carve-up (~117 MB for N=100k)
    float*    h1   = (float*)d_ws;
    float*    agg1 = h1   + nn * 64;   // becomes x after finalize1
    float*    lin2 = agg1 + nn * 64;
    float*    agg2 = lin2 + nn * 64;   // becomes scores after finalize2
    float*    a_s1 = agg2 + nn * 64;
    float*    a_d1 = a_s1 + nn * 8;
    unsigned* m1   = (unsigned*)(a_d1 + nn * 8);
    float*    s1   = (float*)(m1 + nn * 8);
    float*    a_s2 = s1   + nn * 8;
    float*    a_d2 = a_s2 + nn;
    unsigned* m2   = (unsigned*)(a_d2 + nn);
    float*    s2   = (float*)(m2 + nn);
    float*    out  = (float*)d_out;

    // zero accumulators (key 0 == -inf for the encoded max)
    hipMemsetAsync(agg1, 0, nn * 64 * sizeof(float), stream);
    hipMemsetAsync(agg2, 0, nn * 64 * sizeof(float), stream);
    hipMemsetAsync(m1,   0, nn * 8 * sizeof(unsigned), stream);
    hipMemsetAsync(s1,   0, nn * 8 * sizeof(float), stream);
    hipMemsetAsync(m2,   0, nn * sizeof(unsigned), stream);
    hipMemsetAsync(s2,   0, nn * sizeof(float), stream);
    hipMemsetAsync(out,  0, sizeof(float), stream);

    const int T = 256;
    dim3 blk(T);

    // layer 1
    gat_gemm1<<<dim3((n + 127) / 128), blk, 0, stream>>>(feat, W1, h1, n);
    gat_attn1<<<dim3((n * 8 + T - 1) / T), blk, 0, stream>>>(h1, att_s1, att_d1, a_s1, a_d1, n);
    gat_edge_max1<<<dim3((etot * 8 + T - 1) / T), blk, 0, stream>>>(ei, e, etot, a_s1, a_d1, m1);
    gat_edge_agg1<<<dim3((etot + 7) / 8), blk, 0, stream>>>(ei, e, etot, a_s1, a_d1, m1, s1, h1, agg1);
    gat_finalize1<<<dim3((n * 64 + T - 1) / T), blk, 0, stream>>>(agg1, s1, b1, n);

    // layer 2
    gat_gemm2<<<dim3((n + 127) / 128), blk, 0, stream>>>(agg1, W2, lin2, n);
    gat_attn2<<<dim3((n + T - 1) / T), blk, 0, stream>>>(lin2, att_s2, att_d2, a_s2, a_d2, n);
    gat_edge_max2<<<dim3((etot + T - 1) / T), blk, 0, stream>>>(ei, e, etot, a_s2, a_d2, m2);
    gat_edge_agg2<<<dim3((etot + 7) / 8), blk, 0, stream>>>(ei, e, etot, a_s2, a_d2, m2, s2, lin2, agg2);
    gat_finalize2<<<dim3((n * 64 + T - 1) / T), blk, 0, stream>>>(agg2, s2, b2, n);

    // loss / pred / labels
    gat_loss<<<dim3((n + T - 1) / T), blk, 0, stream>>>(agg2, label, out, n);
}